// GraphSAGELayer_3573412790300
// MI455X (gfx1250) — compile-verified
//
#include <hip/hip_runtime.h>
#include <math.h>

typedef __attribute__((ext_vector_type(2))) float v2f;
typedef __attribute__((ext_vector_type(8))) float v8f;
typedef int gvi4 __attribute__((vector_size(16)));   // matches builtin param type

#if defined(__has_builtin)
#  if __has_builtin(__builtin_amdgcn_global_load_async_to_lds_b128)
#    define HAVE_ASYNC_LDS 1
#  endif
#endif

#define NNODES 2048
#define IN_DIM 64
#define OUT_DIM 128
#define TKA 64          // K tile per iteration
#define KHALF 1024      // K range per wave-group (split-K x2)
#define ADJ_S 68        // padded stride (ints) for 16-row adj tile
#define XT_S 68         // padded stride for transposed x tile
#define HS 132          // padded stride for 128-wide LDS tiles (kernel 2)
#define XS2 68          // padded stride for 16x64 x tile (kernel 2)
#define NEG_INF (-3.402823466e38f)

#define ADJ_TILE (16 * ADJ_S)            // 1088 words
#define XT_TILE  (64 * XT_S)             // 4352 words
#define OFF_XT   (4 * ADJ_TILE)          // adj: [2 groups][2 bufs][ADJ_TILE]
#define OFF_DEG  (OFF_XT + 2 * XT_TILE)  // 13056
#define SMEM_F   (OFF_DEG + 32)          // 13088 words = ~52 KB
#define OFF_MEANP OFF_XT                 // combine buffers alias xT (post-loop)
#define OFF_MAXP  (OFF_XT + 1024)

// 16-byte global -> LDS copy: async DMA on CDNA5, sync fallback otherwise.
__device__ __forceinline__ void async_cp16(const int* g, int* l) {
#ifdef HAVE_ASYNC_LDS
    __builtin_amdgcn_global_load_async_to_lds_b128(
        (gvi4*)const_cast<int*>(g), (gvi4*)l, 0, 0);
#else
    *(int4*)l = *(const int4*)g;
#endif
}

__device__ __forceinline__ void wait_async0() {
#ifdef HAVE_ASYNC_LDS
#  if __has_builtin(__builtin_amdgcn_s_wait_asynccnt)
    __builtin_amdgcn_s_wait_asynccnt(0);
#  else
    asm volatile("s_wait_asynccnt 0x0" ::: "memory");
#  endif
#endif
}

// ---------------------------------------------------------------------------
// Kernel 1: fused mean/max aggregation, split-K x2 within the workgroup.
//   8 waves: waves 0-3 -> neighbors [0,1024), waves 4-7 -> [1024,2048).
//   Wave (g, wg) owns a 16x16 tile (16 rows x features 16*wg..16*wg+15).
//   Adjacency staged raw (int) by double-buffered async LDS DMA.
//   x tile staged transposed so max loop reads are b128.
// ---------------------------------------------------------------------------
__global__ __launch_bounds__(256) void sage_aggregate_kernel(
    const float* __restrict__ x,       // [N][64]
    const int*   __restrict__ adj,     // [N][N]
    float*       __restrict__ neigh)   // [N][128]
{
    __shared__ __align__(16) float smem[SMEM_F];
    int*   adjI = (int*)smem;                 // [2][2][ADJ_TILE]
    float* xT   = smem + OFF_XT;              // [2][XT_TILE]
    int*   degS = (int*)(smem + OFF_DEG);     // [2][16]

    const int t    = threadIdx.x;
    const int lane = t & 31;
    const int wave = t >> 5;          // 0..7
    const int g    = wave >> 2;       // split-K group
    const int wg   = wave & 3;        // wave within group
    const int half = lane >> 4;
    const int lq   = lane & 15;
    const int n0   = wg * 16;
    const int tid  = t & 127;         // thread index within group
    const int row0 = blockIdx.x * 16;

    int*   adjG  = adjI + g * 2 * ADJ_TILE;
    float* xTg   = xT + g * XT_TILE;
    const int kbase = g * KHALF;

    v8f acc = {};
    float mx[8];
    #pragma unroll
    for (int r = 0; r < 8; ++r) mx[r] = NEG_INF;
    int deg = 0;

    // prefetch first adjacency tile into buffer 0 (2 x 16B per thread)
    #pragma unroll
    for (int j = 0; j < 2; ++j) {
        const int u = tid + j * 128;          // 256 16B units per tile
        const int r = u >> 4, c = (u & 15) * 4;
        async_cp16(adj + (size_t)(row0 + r) * NNODES + kbase + c,
                   adjG + r * ADJ_S + c);
    }

    for (int it = 0; it < KHALF / TKA; ++it) {
        const int kt  = kbase + it * TKA;
        const int buf = it & 1;
        const int* adjT = adjG + buf * ADJ_TILE;

        // ---- stage x tile transposed: xT[c][r] = x[kt+r][c] ----
        {
            const int r  = tid & 63;
            const int c0 = (tid >> 6) * 32;
            const float4* s4 = (const float4*)(x + (size_t)(kt + r) * IN_DIM + c0);
            #pragma unroll
            for (int i = 0; i < 8; ++i) {
                const float4 v = s4[i];
                const int c = c0 + 4 * i;
                xTg[(c + 0) * XT_S + r] = v.x;
                xTg[(c + 1) * XT_S + r] = v.y;
                xTg[(c + 2) * XT_S + r] = v.z;
                xTg[(c + 3) * XT_S + r] = v.w;
            }
        }
        wait_async0();
        __syncthreads();

        // ---- prefetch next adjacency tile into the other buffer ----
        if (it + 1 < KHALF / TKA) {
            int* nb = adjG + (buf ^ 1) * ADJ_TILE;
            const int nkt = kt + TKA;
            #pragma unroll
            for (int j = 0; j < 2; ++j) {
                const int u = tid + j * 128;
                const int r = u >> 4, c = (u & 15) * 4;
                async_cp16(adj + (size_t)(row0 + r) * NNODES + nkt + c,
                           nb + r * ADJ_S + c);
            }
        }

        // ---- mean partial via f32 WMMA (16x16x4) ----
        #pragma unroll
        for (int kk = 0; kk < TKA; kk += 4) {
            const int kA = kk + 2 * half;
            v2f a, b;
            a.x = (float)adjT[lq * ADJ_S + kA];
            a.y = (float)adjT[lq * ADJ_S + kA + 1];
            b.x = xTg[(n0 + lq) * XT_S + kA];
            b.y = xTg[(n0 + lq) * XT_S + kA + 1];
            acc = __builtin_amdgcn_wmma_f32_16x16x4_f32(
                false, a, false, b, (short)0, acc, false, false);
        }

        // ---- masked max partial, 4 k per step (b128 LDS reads) ----
        for (int k = 0; k < TKA; k += 4) {
            const float4 xv = *(const float4*)&xTg[(n0 + lq) * XT_S + k];
            #pragma unroll
            for (int r = 0; r < 8; ++r) {
                const int4 av = *(const int4*)&adjT[(r + 8 * half) * ADJ_S + k];
                const float m0 = fmaxf(av.x > 0 ? xv.x : NEG_INF,
                                       av.y > 0 ? xv.y : NEG_INF);
                const float m1 = fmaxf(av.z > 0 ? xv.z : NEG_INF,
                                       av.w > 0 ? xv.w : NEG_INF);
                mx[r] = fmaxf(mx[r], fmaxf(m0, m1));
            }
        }

        // ---- degree partial (one lane per row, int adds) ----
        if (tid < 16) {
            #pragma unroll
            for (int k = 0; k < TKA; k += 4) {
                const int4 av = *(const int4*)&adjT[tid * ADJ_S + k];
                deg += av.x + av.y + av.z + av.w;
            }
        }
        __syncthreads();
    }

    // ---- combine split-K partials through LDS ----
    if (tid < 16) degS[g * 16 + tid] = deg;
    float* meanP = smem + OFF_MEANP;
    float* maxP  = smem + OFF_MAXP;
    if (g == 1) {
        #pragma unroll
        for (int r = 0; r < 8; ++r) {
            const int m = r + 8 * half;
            meanP[m * 64 + n0 + lq] = acc[r];
            maxP [m * 64 + n0 + lq] = mx[r];
        }
    }
    __syncthreads();
    if (g == 0) {
        #pragma unroll
        for (int r = 0; r < 8; ++r) {
            const int m = r + 8 * half;
            const float d   = (float)(degS[m] + degS[16 + m]);
            const float sum = acc[r] + meanP[m * 64 + n0 + lq];
            const float mxv = fmaxf(mx[r], maxP[m * 64 + n0 + lq]);
            float* orow = neigh + (size_t)(row0 + m) * OUT_DIM;
            orow[n0 + lq]      = sum / fmaxf(d, 1.0f);
            orow[64 + n0 + lq] = (d > 0.0f) ? mxv : 0.0f;
        }
    }
}

// ---------------------------------------------------------------------------
// Kernel 2: dense head (f32 WMMA GEMMs + GELU + LN).
// ---------------------------------------------------------------------------
__global__ __launch_bounds__(256) void sage_dense_kernel(
    const float* __restrict__ x,      // [N][64]
    const float* __restrict__ neigh,  // [N][128]
    const float* __restrict__ selfW,  // [64][128]
    const float* __restrict__ selfB,  // [128]
    const float* __restrict__ nbW,    // [128][128]
    const float* __restrict__ nbB,    // [128]
    const float* __restrict__ outW,   // [128][128]
    const float* __restrict__ outB,   // [128]
    const float* __restrict__ lnG,    // [128]
    const float* __restrict__ lnB,    // [128]
    float*       __restrict__ out)    // [N][128]
{
    __shared__ float x_t [16 * XS2];
    __shared__ float ng_t[16 * HS];
    __shared__ float h_t [16 * HS];
    __shared__ float o_t [16 * HS];
    __shared__ float mean_s[16], rstd_s[16];

    const int t    = threadIdx.x;
    const int lane = t & 31;
    const int wave = t >> 5;        // 0..7
    const int half = lane >> 4;
    const int lq   = lane & 15;
    const int n0   = wave * 16;
    const int row0 = blockIdx.x * 16;

    {
        const int r  = t >> 4;
        const int c0 = (t & 15) * 4;
        *(float4*)(&x_t[r * XS2 + c0]) =
            *(const float4*)(x + (size_t)(row0 + r) * IN_DIM + c0);
    }
    {
        const int r  = t >> 4;
        const int c0 = (t & 15) * 8;
        const float4* s4 = (const float4*)(neigh + (size_t)(row0 + r) * OUT_DIM + c0);
        *(float4*)(&ng_t[r * HS + c0])     = s4[0];
        *(float4*)(&ng_t[r * HS + c0 + 4]) = s4[1];
    }
    __syncthreads();

    v8f h = {};
    #pragma unroll
    for (int k = 0; k < IN_DIM; k += 4) {
        const int kA = k + 2 * half;
        v2f a, b;
        a.x = x_t[lq * XS2 + kA];
        a.y = x_t[lq * XS2 + kA + 1];
        b.x = selfW[kA * OUT_DIM + n0 + lq];
        b.y = selfW[(kA + 1) * OUT_DIM + n0 + lq];
        h = __builtin_amdgcn_wmma_f32_16x16x4_f32(false, a, false, b, (short)0, h, false, false);
    }
    #pragma unroll
    for (int k = 0; k < OUT_DIM; k += 4) {
        const int kA = k + 2 * half;
        v2f a, b;
        a.x = ng_t[lq * HS + kA];
        a.y = ng_t[lq * HS + kA + 1];
        b.x = nbW[kA * OUT_DIM + n0 + lq];
        b.y = nbW[(kA + 1) * OUT_DIM + n0 + lq];
        h = __builtin_amdgcn_wmma_f32_16x16x4_f32(false, a, false, b, (short)0, h, false, false);
    }
    {
        const float bias = selfB[n0 + lq] + nbB[n0 + lq];
        #pragma unroll
        for (int r = 0; r < 8; ++r)
            h_t[(r + 8 * half) * HS + n0 + lq] = h[r] + bias;
    }
    __syncthreads();

    v8f o = {};
    #pragma unroll
    for (int k = 0; k < OUT_DIM; k += 4) {
        const int kA = k + 2 * half;
        v2f a, b;
        a.x = h_t[lq * HS + kA];
        a.y = h_t[lq * HS + kA + 1];
        b.x = outW[kA * OUT_DIM + n0 + lq];
        b.y = outW[(kA + 1) * OUT_DIM + n0 + lq];
        o = __builtin_amdgcn_wmma_f32_16x16x4_f32(false, a, false, b, (short)0, o, false, false);
    }
    {
        const float ob = outB[n0 + lq];
        #pragma unroll
        for (int r = 0; r < 8; ++r) {
            float v = o[r] + ob;
            v = 0.5f * v * (1.0f + erff(v * 0.70710678118654752f));  // exact GELU
            o_t[(r + 8 * half) * HS + n0 + lq] = v;
        }
    }
    __syncthreads();

    if (t < 16) {
        float s = 0.0f, ss = 0.0f;
        #pragma unroll 8
        for (int c = 0; c < OUT_DIM; ++c) {
            const float v = o_t[t * HS + c];
            s += v;
            ss = fmaf(v, v, ss);
        }
        const float m   = s * (1.0f / OUT_DIM);
        const float var = ss * (1.0f / OUT_DIM) - m * m;
        mean_s[t] = m;
        rstd_s[t] = rsqrtf(var + 1e-5f);
    }
    __syncthreads();

    {
        const int r  = t >> 4;
        const int c0 = (t & 15) * 8;
        const float m  = mean_s[r];
        const float rs = rstd_s[r];
        float res[8];
        #pragma unroll
        for (int i = 0; i < 8; ++i) {
            const int c = c0 + i;
            const float v = o_t[r * HS + c];
            res[i] = (v - m) * rs * lnG[c] + lnB[c];
        }
        float* dst = out + (size_t)(row0 + r) * OUT_DIM + c0;
        *(float4*)(dst)     = make_float4(res[0], res[1], res[2], res[3]);
        *(float4*)(dst + 4) = make_float4(res[4], res[5], res[6], res[7]);
    }
}

extern "C" void kernel_launch(void* const* d_in, const int* in_sizes, int n_in,
                              void* d_out, int out_size, void* d_ws, size_t ws_size,
                              hipStream_t stream) {
    const float* x     = (const float*)d_in[0];
    const int*   adj   = (const int*)  d_in[1];
    const float* selfW = (const float*)d_in[2];
    const float* selfB = (const float*)d_in[3];
    const float* nbW   = (const float*)d_in[4];
    const float* nbB   = (const float*)d_in[5];
    const float* outW  = (const float*)d_in[6];
    const float* outB  = (const float*)d_in[7];
    const float* lnG   = (const float*)d_in[8];
    const float* lnB   = (const float*)d_in[9];
    float* outp  = (float*)d_out;
    float* neigh = (float*)d_ws;   // 2048*128 floats = 1 MB scratch

    sage_aggregate_kernel<<<NNODES / 16, 256, 0, stream>>>(x, adj, neigh);
    sage_dense_kernel<<<NNODES / 16, 256, 0, stream>>>(
        x, neigh, selfW, selfB, nbW, nbB, outW, outB, lnG, lnB, outp);
}